// MultiScaleGraphConv_31361851195614
// MI455X (gfx1250) — compile-verified
//
#include <hip/hip_runtime.h>
#include <math.h>

// ---------------------------------------------------------------------------
// MultiScaleGraphConv for MI455X (gfx1250, wave32).
//  - GEMMs via V_WMMA_F32_16X16X4_F32 (f32 in/out: precision-preserving; the
//    workload is bound by ~12 GB of random edge traffic, not matmul FLOPs).
//  - A-tile staged in LDS (padded) -> ds_load traffic, coalesced global loads.
//  - Edge scatter: 64 lanes/edge, float4 gather + global_atomic_add_f32.
// ---------------------------------------------------------------------------

#define NN 50000
#define NE 800000
#define IND 256
#define OUTD 768

typedef __attribute__((ext_vector_type(2))) float v2f;
typedef __attribute__((ext_vector_type(8))) float v8f;

// ---------------- degree / norm ----------------
__global__ void k_deg_init(float* deg) {
  int i = blockIdx.x * blockDim.x + threadIdx.x;
  if (i < NN) deg[i] = 1.0f;                 // self-loop weight 1 counts toward deg
}

__global__ void k_deg_edges(const long long* __restrict__ ei,
                            const float* __restrict__ ew,
                            float* __restrict__ deg) {
  int e = blockIdx.x * blockDim.x + threadIdx.x;
  if (e < NE) {
    int d = (int)ei[NE + e];                 // row 1 = dst
    atomicAdd(&deg[d], ew[e]);
  }
}

__global__ void k_dinv(float* deg) {
  int i = blockIdx.x * blockDim.x + threadIdx.x;
  if (i < NN) {
    float d = deg[i];
    deg[i] = (d > 0.0f) ? rsqrtf(d) : 0.0f;  // in-place deg -> dinv
  }
}

// ---------------- GEMM: C[M x OUTC] = A[M x K] @ W[K x OUTC] ---------------
// block = 256 threads = 8 waves; block tile = 16 rows x 128 cols.
// wave tile = 16x16 via v_wmma_f32_16x16x4_f32 (K stepped by 4).
// A tile (16 x K) staged in LDS with padded stride (K+4) to avoid bank clash.
__global__ void k_gemm_wmma(const float* __restrict__ A, int lda,
                            const float* __restrict__ W,   // K x OUTC row-major
                            float* __restrict__ C, int ldc,
                            int K, int OUTC) {
  extern __shared__ float sA[];              // 16 * (K+4) floats
  const int lane = threadIdx.x & 31;
  const int wave = threadIdx.x >> 5;
  const int row0 = blockIdx.x * 16;
  const int col0 = blockIdx.y * 128 + wave * 16;
  const int sstride = K + 4;

  // cooperative load of the 16 x K A tile (coalesced float4)
  const int k4 = K >> 2;
  for (int i = threadIdx.x; i < 16 * k4; i += 256) {
    int r = i / k4, c = (i - r * k4) << 2;
    *(float4*)&sA[r * sstride + c] =
        *(const float4*)&A[(size_t)(row0 + r) * lda + c];
  }
  __syncthreads();

  const int m  = lane & 15;                  // A row / C col index within tile
  const int kh = (lane >> 4) << 1;           // 0 or 2 (K half select)
  const float* sArow = &sA[m * sstride + kh];
  const float* bcol  = W + (size_t)kh * OUTC + (col0 + m);

  v8f c = {0.f, 0.f, 0.f, 0.f, 0.f, 0.f, 0.f, 0.f};
  #pragma unroll 4
  for (int k = 0; k < K; k += 4) {
    v2f a, b;
    a[0] = sArow[k];                         // A[m][k+kh]
    a[1] = sArow[k + 1];                     // A[m][k+kh+1]
    b[0] = bcol[(size_t)k * OUTC];           // W[k+kh][col]
    b[1] = bcol[(size_t)(k + 1) * OUTC];     // W[k+kh+1][col]
    c = __builtin_amdgcn_wmma_f32_16x16x4_f32(false, a, false, b,
                                              (short)0, c, false, false);
  }

  // C/D layout: lane(0-15)->M=v, lane(16-31)->M=8+v; N = lane&15
  const int half = lane >> 4;
  #pragma unroll
  for (int v = 0; v < 8; ++v)
    C[(size_t)(row0 + half * 8 + v) * ldc + col0 + m] = c[v];
}

// ---------------- aggregation: self-loop init (+bias), then edge scatter ---
__global__ void k_agg_init(const float* __restrict__ h,
                           const float* __restrict__ dinv,
                           const float* __restrict__ bias,
                           float* __restrict__ out, int ldo) {
  int t = blockIdx.x * blockDim.x + threadIdx.x;   // NN * 64 groups
  int node = t >> 6;
  int g = (t & 63) << 2;
  if (node < NN) {
    float s = dinv[node]; s = s * s;                // norm of self loop
    float4 hv = *(const float4*)(h + (size_t)node * IND + g);
    float4 o;
    o.x = s * hv.x + bias[g + 0];
    o.y = s * hv.y + bias[g + 1];
    o.z = s * hv.z + bias[g + 2];
    o.w = s * hv.w + bias[g + 3];
    *(float4*)(out + (size_t)node * ldo + g) = o;
  }
}

__global__ void k_scatter(const float* __restrict__ h,
                          const long long* __restrict__ ei,
                          const float* __restrict__ ew,
                          const float* __restrict__ dinv,
                          float* __restrict__ out, int ldo) {
  long long t = (long long)blockIdx.x * blockDim.x + threadIdx.x;
  int e = (int)(t >> 6);
  int g = ((int)t & 63) << 2;
  if (e < NE) {
    int s = (int)ei[e];
    int d = (int)ei[NE + e];
    float nrm = dinv[s] * ew[e] * dinv[d];
    float4 hv = *(const float4*)(h + (size_t)s * IND + g);
    float* o = out + (size_t)d * ldo + g;
    atomicAdd(o + 0, nrm * hv.x);
    atomicAdd(o + 1, nrm * hv.y);
    atomicAdd(o + 2, nrm * hv.z);
    atomicAdd(o + 3, nrm * hv.w);
  }
}

// ---------------- fused bias + LayerNorm + exact GELU (in place) -----------
__global__ void k_ln_gelu(float* __restrict__ out,
                          const float* __restrict__ bp,
                          const float* __restrict__ gamma,
                          const float* __restrict__ beta) {
  __shared__ float s_sum[256];
  __shared__ float s_sq[256];
  int row = blockIdx.x;
  float* p = out + (size_t)row * OUTD;
  float v[3], lsum = 0.f, lsq = 0.f;
  #pragma unroll
  for (int i = 0; i < 3; ++i) {
    int d = threadIdx.x + i * 256;
    float tv = p[d] + bp[d];
    v[i] = tv; lsum += tv; lsq += tv * tv;
  }
  s_sum[threadIdx.x] = lsum; s_sq[threadIdx.x] = lsq;
  __syncthreads();
  for (int off = 128; off > 0; off >>= 1) {
    if (threadIdx.x < off) {
      s_sum[threadIdx.x] += s_sum[threadIdx.x + off];
      s_sq[threadIdx.x]  += s_sq[threadIdx.x + off];
    }
    __syncthreads();
  }
  float mean = s_sum[0] * (1.0f / OUTD);
  float var  = s_sq[0] * (1.0f / OUTD) - mean * mean;
  float rstd = rsqrtf(var + 1e-5f);
  #pragma unroll
  for (int i = 0; i < 3; ++i) {
    int d = threadIdx.x + i * 256;
    float y = (v[i] - mean) * rstd * gamma[d] + beta[d];
    p[d] = 0.5f * y * (1.0f + erff(y * 0.70710678118654752440f));
  }
}

// ---------------------------------------------------------------------------
extern "C" void kernel_launch(void* const* d_in, const int* in_sizes, int n_in,
                              void* d_out, int out_size, void* d_ws, size_t ws_size,
                              hipStream_t stream) {
  const float*     x     = (const float*)d_in[0];
  const long long* ei    = (const long long*)d_in[1];   // int64 [2, E]
  const float*     ea    = (const float*)d_in[2];
  const float*     W1    = (const float*)d_in[3];
  const float*     b1    = (const float*)d_in[4];
  const float*     W2    = (const float*)d_in[5];
  const float*     b2    = (const float*)d_in[6];
  const float*     W3    = (const float*)d_in[7];
  const float*     b3    = (const float*)d_in[8];
  const float*     Wp    = (const float*)d_in[9];
  const float*     bp    = (const float*)d_in[10];
  const float*     gamma = (const float*)d_in[11];
  const float*     beta  = (const float*)d_in[12];
  float* out = (float*)d_out;

  // workspace layout (floats): dinv[NN] | hbuf[NN*256] | cat[NN*768]
  float* ws   = (float*)d_ws;
  float* dinv = ws;
  float* hbuf = ws + NN;                       // NN*4 bytes % 16 == 0
  float* cat  = hbuf + (size_t)NN * IND;

  // 1) degree + inverse-sqrt normalization
  k_deg_init <<<(NN + 255) / 256, 256, 0, stream>>>(dinv);
  k_deg_edges<<<(NE + 255) / 256, 256, 0, stream>>>(ei, ea, dinv);
  k_dinv     <<<(NN + 255) / 256, 256, 0, stream>>>(dinv);

  const float* Ws[3] = {W1, W2, W3};
  const float* bs[3] = {b1, b2, b3};
  const int scales[3] = {1, 2, 4};

  // 2) three branches; intermediate features live inside `cat` columns
  for (int br = 0; br < 3; ++br) {
    const float* fin = x;
    int lda = IND;
    float* fout = cat + br * IND;              // column block, stride OUTD
    for (int it = 0; it < scales[br]; ++it) {
      dim3 g(NN / 16, IND / 128);
      size_t sh = 16 * (size_t)(IND + 4) * sizeof(float);
      k_gemm_wmma<<<g, 256, sh, stream>>>(fin, lda, Ws[br], hbuf, IND, IND, IND);
      k_agg_init <<<((size_t)NN * 64 + 255) / 256, 256, 0, stream>>>(
          hbuf, dinv, bs[br], fout, OUTD);
      k_scatter  <<<((size_t)NE * 64 + 255) / 256, 256, 0, stream>>>(
          hbuf, ei, ea, dinv, fout, OUTD);
      fin = fout; lda = OUTD;
    }
  }

  // 3) projection GEMM -> d_out, then fused bias+LN+GELU in place
  {
    dim3 g(NN / 16, OUTD / 128);
    size_t sh = 16 * (size_t)(OUTD + 4) * sizeof(float);
    k_gemm_wmma<<<g, 256, sh, stream>>>(cat, OUTD, Wp, out, OUTD, OUTD, OUTD);
  }
  k_ln_gelu<<<NN, 256, 0, stream>>>(out, bp, gamma, beta);
}